// UV_Aggregator_48765058678794
// MI455X (gfx1250) — compile-verified
//
#include <hip/hip_runtime.h>
#include <math.h>

// ---------------------------------------------------------------------------
// UV_Aggregator fused kernels for MI455X (gfx1250, wave32)
//   N=4096 nodes, H=200 user-hist, A=50 item-adj, L=250, D=64, k=125
// Kernel 1: query = [self|target] @ W_lin^T + b  via V_WMMA_F32_16X16X4_F32
// Kernel 2: per-node fused gather (async direct-to-LDS) -> dist+gumbel key ->
//           rank top-k -> attention softmax -> weighted sum.
// This op is gather-bound (262 MB of L2-resident embedding reads, compute is
// ~0.5 GFLOP): the wins are (a) touching every embedding row exactly once,
// staged in gfx1250's 320KB-per-WGP LDS, and (b) using the CDNA5 async
// memory->LDS path (GLOBAL_LOAD_ASYNC_TO_LDS_B128, ASYNCcnt) so the gather
// never round-trips through VGPRs.
// ---------------------------------------------------------------------------

typedef float v2f __attribute__((ext_vector_type(2)));
typedef float v8f __attribute__((ext_vector_type(8)));

#define DIM   64
#define KDIM  128      // 2*D
#define HN    200
#define AN    50
#define LN    250
#define NREL  8
#define ESTRIDE 68     // LDS row stride in floats: 16B-aligned, bank-spread

// ---------------------------------------------------------------------------
// Kernel 1: C(N x 64) = X(N x 128) @ W^T(128 x 64) + b,  X = [self | target]
// One wave per 16x16 output tile, K marched in steps of 4 (f32 WMMA).
// A-frag layout (ISA 7.12.2, 32-bit A 16x4): lanes 0-15 hold M=lane,K={k0,k0+1};
// lanes 16-31 hold M=lane-16, K={k0+2,k0+3}. B mirrors (row=K striped on lanes).
// C/D: VGPR v -> row v (lanes 0-15) / row v+8 (lanes 16-31), col = lane&15.
// ---------------------------------------------------------------------------
__global__ __launch_bounds__(128)
void query_gemm_wmma(const float* __restrict__ selfF,
                     const float* __restrict__ targF,
                     const float* __restrict__ W,     // (64,128) row-major
                     const float* __restrict__ bvec,  // (64)
                     float* __restrict__ q,           // (N,64)
                     int N)
{
  const int lane = threadIdx.x & 31;
  const int wv   = threadIdx.x >> 5;     // wave id 0..3 -> column tile
  const int m0   = blockIdx.x * 16;
  const int c0   = wv * 16;
  const int lm   = lane & 15;
  const int hi   = lane >> 4;            // 0 or 1
  const int row  = m0 + lm;
  const int col  = c0 + lm;

  v8f acc = {};
  #pragma unroll
  for (int k0 = 0; k0 < KDIM; k0 += 4) {
    const int kk = k0 + 2 * hi;          // kk even, kk+1 in same half-matrix
    v2f a, b;
    if (kk < DIM) {
      a.x = selfF[row * DIM + kk];
      a.y = selfF[row * DIM + kk + 1];
    } else {
      a.x = targF[row * DIM + kk - DIM];
      a.y = targF[row * DIM + kk + 1 - DIM];
    }
    b.x = W[col * KDIM + kk];            // B[k][col] = W[col][k]
    b.y = W[col * KDIM + kk + 1];
    acc = __builtin_amdgcn_wmma_f32_16x16x4_f32(false, a, false, b,
                                                (short)0, acc, false, false);
  }

  const float bias = bvec[col];
  #pragma unroll
  for (int v = 0; v < 8; ++v) {
    const int r = m0 + v + hi * 8;
    q[r * DIM + col] = acc[v] + bias;
  }
}

// ---------------------------------------------------------------------------
// Kernel 2: one workgroup (256 thr = 8 waves) per node. ~76 KB LDS/WG ->
// 4 workgroups per WGP (320 KB LDS budget on CDNA5).
// ---------------------------------------------------------------------------
__global__ __launch_bounds__(256)
void uv_agg_fused(const float* __restrict__ q,        // (N,64) workspace
                  const int*   __restrict__ hist_uv,  // (N,200)
                  const int*   __restrict__ adj,      // (N,50)
                  const int*   __restrict__ labels,   // (N,250)
                  const float* __restrict__ gum,      // (N,250)
                  const float* __restrict__ u2e,      // (100000,64)
                  const float* __restrict__ v2e,      // (100000,64)
                  const float* __restrict__ r2e,      // (8,64)
                  const float* __restrict__ att,      // (128)
                  const int*   __restrict__ kptr,     // scalar k (=125)
                  float* __restrict__ out)            // (N,64)
{
  const int tid = threadIdx.x;
  const int n   = blockIdx.x;
  const int K   = kptr[0];

  __shared__ __align__(16) float e_lds[LN * ESTRIDE]; // 68000 B neighbor tile
  __shared__ float q_lds[DIM];
  __shared__ float att_lds[KDIM];
  __shared__ float r2e_lds[NREL * DIM];
  __shared__ int   lab_lds[256];
  __shared__ float key_lds[256];
  __shared__ float red_lds[256];
  __shared__ float w_lds[256];
  __shared__ float part_lds[256];

  // ---- stage small operands ----
  if (tid < DIM)                 q_lds[tid]        = q[(size_t)n * DIM + tid];
  if (tid >= 64 && tid < 192)    att_lds[tid - 64] = att[tid - 64];
  for (int i = tid; i < NREL * DIM; i += 256) r2e_lds[i] = r2e[i];
  if (tid < LN)                  lab_lds[tid]      = labels[(size_t)n * LN + tid];

  // ---- gather 250 neighbor embedding rows DIRECTLY into LDS via the CDNA5
  //      async memory->LDS path (no VGPR round trip, tracked by ASYNCcnt).
  //      250 rows * 16 float4 chunks; 16 consecutive threads share one row
  //      index (broadcast index read, 256B coalesced row read from L2). ----
  for (int c = tid; c < LN * 16; c += 256) {
    const int i = c >> 4;
    const int j = c & 15;
    const int rowIdx = (i < HN) ? hist_uv[(size_t)n * HN + i]
                                : adj[(size_t)n * AN + (i - HN)];
    const float* src = ((i < HN) ? u2e : v2e) + (size_t)rowIdx * DIM + 4 * j;
    // low 32 bits of a flat pointer into __shared__ == LDS byte address
    // (ISA 10.2 aperture mapping), which is what the async op's VDST takes.
    const unsigned ldsAddr =
        (unsigned)(uintptr_t)(&e_lds[i * ESTRIDE + 4 * j]);
    asm volatile("global_load_async_to_lds_b128 %0, %1, off"
                 :: "v"(ldsAddr), "v"(src)
                 : "memory");
  }
  // wave-local completion of all async transfers, then cross-wave visibility
  asm volatile("s_wait_asynccnt 0x0" ::: "memory");
  __syncthreads();

  // ---- selection key: -||q - e_i|| + gumbel(u_i)  (log_softmax shift is a
  //      per-node constant -> drops out of top-k) ----
  if (tid < LN) {
    float s = 0.f;
    #pragma unroll 8
    for (int d = 0; d < DIM; ++d) {
      const float diff = q_lds[d] - e_lds[tid * ESTRIDE + d];
      s = fmaf(diff, diff, s);
    }
    const float u = gum[(size_t)n * LN + tid];
    const float g = -logf(-logf(u + 1e-10f) + 1e-10f);
    key_lds[tid] = g - sqrtf(s);
  }
  __syncthreads();

  // ---- top-K set via rank counting (stable tie-break = JAX top_k order;
  //      downstream is permutation-invariant so the set is all we need) ----
  bool sel = false;
  if (tid < LN) {
    const float ki = key_lds[tid];
    int r = 0;
    for (int j = 0; j < LN; ++j) {         // LDS broadcast reads
      const float kj = key_lds[j];
      r += (kj > ki) || (kj == ki && j < tid);
    }
    sel = (r < K);
  }

  // ---- attention score for selected neighbors ----
  float sc = 0.f;
  if (sel) {
    const int l = lab_lds[tid];
    #pragma unroll 8
    for (int d = 0; d < DIM; ++d) {
      sc = fmaf(e_lds[tid * ESTRIDE + d], att_lds[d],       sc);
      sc = fmaf(r2e_lds[l * DIM + d],     att_lds[DIM + d], sc);
    }
  }

  // ---- softmax over the selected set (block tree reductions) ----
  red_lds[tid] = sel ? sc : -3.0e38f;
  __syncthreads();
  for (int s = 128; s > 0; s >>= 1) {
    if (tid < s) red_lds[tid] = fmaxf(red_lds[tid], red_lds[tid + s]);
    __syncthreads();
  }
  const float mx = red_lds[0];
  __syncthreads();
  const float ex = sel ? expf(sc - mx) : 0.f;
  red_lds[tid] = ex;
  __syncthreads();
  for (int s = 128; s > 0; s >>= 1) {
    if (tid < s) red_lds[tid] += red_lds[tid + s];
    __syncthreads();
  }
  const float denom = red_lds[0];
  w_lds[tid] = ex / denom;                 // 0 for unselected lanes
  __syncthreads();

  // ---- out[d] = sum_i w_i * e[i][d]; 4 row-groups x 64 cols, then reduce ----
  const int d   = tid & 63;
  const int grp = tid >> 6;
  float acc = 0.f;
  for (int i = grp; i < LN; i += 4)
    acc = fmaf(w_lds[i], e_lds[i * ESTRIDE + d], acc);
  part_lds[tid] = acc;
  __syncthreads();
  if (tid < DIM)
    out[(size_t)n * DIM + tid] = part_lds[tid] + part_lds[64 + tid] +
                                 part_lds[128 + tid] + part_lds[192 + tid];
}

// ---------------------------------------------------------------------------
extern "C" void kernel_launch(void* const* d_in, const int* in_sizes, int n_in,
                              void* d_out, int out_size, void* d_ws, size_t ws_size,
                              hipStream_t stream) {
  const float* selfF  = (const float*)d_in[0];
  const float* targF  = (const float*)d_in[1];
  const int*   histUV = (const int*)  d_in[2];
  const int*   adj    = (const int*)  d_in[3];
  const int*   labels = (const int*)  d_in[4];
  const float* gum    = (const float*)d_in[5];
  const float* u2e    = (const float*)d_in[6];
  const float* v2e    = (const float*)d_in[7];
  const float* r2e    = (const float*)d_in[8];
  const float* att    = (const float*)d_in[9];
  const float* Wlin   = (const float*)d_in[10];
  const float* blin   = (const float*)d_in[11];
  const int*   kptr   = (const int*)  d_in[12];

  const int N = in_sizes[0] / DIM;          // 4096
  float* qws = (float*)d_ws;                // N*64 floats = 1 MB scratch

  // Kernel 1: WMMA query GEMM (one 16x16 tile per wave, 4 waves/block)
  query_gemm_wmma<<<N / 16, 128, 0, stream>>>(selfF, targF, Wlin, blin, qws, N);

  // Kernel 2: fully fused per-node aggregation (one workgroup per node)
  uv_agg_fused<<<N, 256, 0, stream>>>(qws, histUV, adj, labels, gum,
                                      u2e, v2e, r2e, att, kptr, (float*)d_out);
}